// BidirectionalSAGEConv_19610820673955
// MI455X (gfx1250) — compile-verified
//
#include <hip/hip_runtime.h>
#include <hip/hip_bf16.h>

// ---------------------------------------------------------------------------
// Bidirectional SAGEConv for MI455X (gfx1250, wave32, WMMA).
// N=100000, E=600000, D=128.
//
// Pipeline (all on `stream`, graph-capture safe):
//   1) memsetAsync: zero sum_f|sum_b|cnt_f|cnt_b
//   2) sage_prep:   w_sum = wr_f + wr_b ; bias_sum = bl_f + bl_b
//   3) sage_edge_agg: one wave per edge, float4/lane, f32 global atomics,
//                     both directions in a single pass over edge_index
//   4) sage_wmma_gemm: fused (sum_f*inv_cnt_f)@wl_f.T + (sum_b*inv_cnt_b)@wl_b.T
//                      + x@w_sum.T + bias, *0.5, relu — mean division fused
//                      into the A-fragment load (per-lane row scale).
// ---------------------------------------------------------------------------

#define NNODES 100000
#define NEDGES 600000
#define DIM    128

typedef __attribute__((ext_vector_type(2))) float v2f;
typedef __attribute__((ext_vector_type(8))) float v8f;

// Workspace layout (floats):
#define OFF_SUMF 0
#define OFF_SUMB ((size_t)NNODES * DIM)
#define OFF_CNTF ((size_t)2 * NNODES * DIM)
#define OFF_CNTB (OFF_CNTF + NNODES)
#define OFF_WSUM (OFF_CNTB + NNODES)
#define OFF_BIAS (OFF_WSUM + DIM * DIM)

// --- prep: w_sum = wr_f + wr_b ; bias_sum = bl_f + bl_b ---------------------
__global__ void sage_prep(const float* __restrict__ wr_f,
                          const float* __restrict__ wr_b,
                          const float* __restrict__ bl_f,
                          const float* __restrict__ bl_b,
                          float* __restrict__ w_sum,
                          float* __restrict__ bias_sum) {
    int idx = blockIdx.x * blockDim.x + threadIdx.x;
    if (idx < DIM * DIM) w_sum[idx] = wr_f[idx] + wr_b[idx];
    if (idx < DIM)       bias_sum[idx] = bl_f[idx] + bl_b[idx];
}

// --- edge aggregation: one wave32 per edge, float4 per lane -----------------
// forward conv:  sum_f[dst] += x[src], cnt_f[dst]++
// backward conv: sum_b[src] += x[dst], cnt_b[src]++   (== flipped edge list)
__global__ __launch_bounds__(256)
void sage_edge_agg(const float* __restrict__ x,
                   const long long* __restrict__ edge_index, // [2][E] int64
                   float* __restrict__ sum_f, float* __restrict__ sum_b,
                   float* __restrict__ cnt_f, float* __restrict__ cnt_b) {
    long long t = (long long)blockIdx.x * blockDim.x + threadIdx.x;
    int e    = (int)(t >> 5);
    int lane = (int)(t & 31);
    if (e >= NEDGES) return;

    int s = (int)edge_index[e];
    int d = (int)edge_index[NEDGES + e];

    const float4 xs = *(const float4*)(x + (size_t)s * DIM + lane * 4);
    const float4 xd = *(const float4*)(x + (size_t)d * DIM + lane * 4);

    float* pf = sum_f + (size_t)d * DIM + lane * 4;
    float* pb = sum_b + (size_t)s * DIM + lane * 4;
    atomicAdd(pf + 0, xs.x); atomicAdd(pf + 1, xs.y);
    atomicAdd(pf + 2, xs.z); atomicAdd(pf + 3, xs.w);
    atomicAdd(pb + 0, xd.x); atomicAdd(pb + 1, xd.y);
    atomicAdd(pb + 2, xd.z); atomicAdd(pb + 3, xd.w);

    if (lane == 0) {
        atomicAdd(cnt_f + d, 1.0f);
        atomicAdd(cnt_b + s, 1.0f);
    }
}

// --- fused triple GEMM + mean-scale + bias + relu (V_WMMA_F32_16X16X4_F32) --
// out[n][o] = relu(0.5*((sum_f[n]/cf[n])@wl_f.T + (sum_b[n]/cb[n])@wl_b.T
//                       + x@w_sum.T + bias)[n][o])
// Block b: node rows [16b, 16b+16). Wave w (of 8): output cols [16w, 16w+16).
// A frag (16x4 f32): lane holds float2 of row (lane&15), K = kk + 2*(lane>>4).
//   -> the mean scale 1/max(cnt,1) is constant per lane; fused as v2f*scalar.
// B frag (4x16 f32) = W.T tile: W row-major [out][in] makes the lane fragment
//   the contiguous float2 W[col0 + (lane&15)][kk + 2*(lane>>4)].
__global__ __launch_bounds__(256)
void sage_wmma_gemm(const float* __restrict__ sum_f,
                    const float* __restrict__ sum_b,
                    const float* __restrict__ cnt_f,
                    const float* __restrict__ cnt_b,
                    const float* __restrict__ x,
                    const float* __restrict__ wl_f,
                    const float* __restrict__ wl_b,
                    const float* __restrict__ w_sum,
                    const float* __restrict__ bias_sum,
                    float* __restrict__ out) {
    const int wave = threadIdx.x >> 5;   // 0..7 -> column tile
    const int lane = threadIdx.x & 31;
    const int r    = lane & 15;
    const int half = lane >> 4;
    const size_t row0 = (size_t)blockIdx.x * 16;
    const int    col0 = wave * 16;

    // per-lane mean reciprocals (row = row0 + r is fixed for this lane)
    const float scale_f = 1.0f / fmaxf(cnt_f[row0 + r], 1.0f);
    const float scale_b = 1.0f / fmaxf(cnt_b[row0 + r], 1.0f);

    const float* aF = sum_f + (row0 + r) * DIM + 2 * half;
    const float* aB = sum_b + (row0 + r) * DIM + 2 * half;
    const float* aX = x     + (row0 + r) * DIM + 2 * half;
    const float* bF = wl_f  + (size_t)(col0 + r) * DIM + 2 * half;
    const float* bB = wl_b  + (size_t)(col0 + r) * DIM + 2 * half;
    const float* bX = w_sum + (size_t)(col0 + r) * DIM + 2 * half;

    v8f acc = {};
#pragma unroll 8
    for (int kk = 0; kk < DIM; kk += 4) {
        v2f a0 = *(const v2f*)(aF + kk) * scale_f;   // fused mean division
        v2f b0 = *(const v2f*)(bF + kk);
        acc = __builtin_amdgcn_wmma_f32_16x16x4_f32(
            false, a0, false, b0, (short)0, acc, false, false);
        v2f a1 = *(const v2f*)(aB + kk) * scale_b;   // fused mean division
        v2f b1 = *(const v2f*)(bB + kk);
        acc = __builtin_amdgcn_wmma_f32_16x16x4_f32(
            false, a1, false, b1, (short)0, acc, false, false);
        v2f a2 = *(const v2f*)(aX + kk);
        v2f b2 = *(const v2f*)(bX + kk);
        acc = __builtin_amdgcn_wmma_f32_16x16x4_f32(
            false, a2, false, b2, (short)0, acc, false, false);
    }

    // C/D layout: col = col0 + (lane&15); VGPR v -> row = row0 + v + 8*(lane>>4)
    const float bias = bias_sum[col0 + r];
    float* op = out + (row0 + (size_t)half * 8) * DIM + col0 + r;
#pragma unroll
    for (int v = 0; v < 8; ++v) {
        float val = 0.5f * (acc[v] + bias);
        op[(size_t)v * DIM] = val > 0.0f ? val : 0.0f;
    }
}

extern "C" void kernel_launch(void* const* d_in, const int* in_sizes, int n_in,
                              void* d_out, int out_size, void* d_ws, size_t ws_size,
                              hipStream_t stream) {
    const float*     x  = (const float*)d_in[0];
    const long long* ei = (const long long*)d_in[1];   // int64 [2][E]
    // d_in[2] = reverse_edge_index (unused: derived from ei)
    const float* wl_f = (const float*)d_in[3];
    const float* bl_f = (const float*)d_in[4];
    const float* wr_f = (const float*)d_in[5];
    const float* wl_b = (const float*)d_in[6];
    const float* bl_b = (const float*)d_in[7];
    const float* wr_b = (const float*)d_in[8];
    float* out = (float*)d_out;

    float* ws       = (float*)d_ws;
    float* sum_f    = ws + OFF_SUMF;
    float* sum_b    = ws + OFF_SUMB;
    float* cnt_f    = ws + OFF_CNTF;
    float* cnt_b    = ws + OFF_CNTB;
    float* w_sum    = ws + OFF_WSUM;
    float* bias_sum = ws + OFF_BIAS;

    // 1) zero accumulators (sums + counts are contiguous)
    hipMemsetAsync(ws, 0,
                   (size_t)(2 * (size_t)NNODES * DIM + 2 * NNODES) * sizeof(float),
                   stream);

    // 2) combined right-weight / bias
    sage_prep<<<(DIM * DIM + 255) / 256, 256, 0, stream>>>(wr_f, wr_b, bl_f, bl_b,
                                                           w_sum, bias_sum);

    // 3) edge aggregation: one wave per edge
    {
        long long total = (long long)NEDGES * 32;
        int blocks = (int)((total + 255) / 256);
        sage_edge_agg<<<blocks, 256, 0, stream>>>(x, ei, sum_f, sum_b, cnt_f, cnt_b);
    }

    // 4) fused mean + triple WMMA GEMM + epilogue (N is exactly 6250 * 16)
    sage_wmma_gemm<<<NNODES / 16, 256, 0, stream>>>(sum_f, sum_b, cnt_f, cnt_b, x,
                                                    wl_f, wl_b, w_sum, bias_sum, out);
}